// GraphConv_996432412685
// MI455X (gfx1250) — compile-verified
//
#include <hip/hip_runtime.h>
#include <math.h>

typedef __attribute__((ext_vector_type(2))) float v2f;
typedef __attribute__((ext_vector_type(8))) float v8f;

#define N_USERS    50000
#define N_ENTITIES 100000
#define CH         64
#define N_EDGES    3200000
#define NNZ        2500000
#define N_HOPS     2

// ---------------------------------------------------------------------------
// init helpers (graph-capture safe: kernels only)
// ---------------------------------------------------------------------------
__global__ void copy2_f4(const float4* __restrict__ src,
                         float4* __restrict__ d1,
                         float4* __restrict__ d2, int n4) {
    int i = blockIdx.x * blockDim.x + threadIdx.x;
    if (i < n4) { float4 v = src[i]; d1[i] = v; d2[i] = v; }
}

__global__ void copy_f4(const float4* __restrict__ src,
                        float4* __restrict__ dst, int n4) {
    int i = blockIdx.x * blockDim.x + threadIdx.x;
    if (i < n4) dst[i] = src[i];
}

__global__ void zero_f4(float4* __restrict__ dst, int n4) {
    int i = blockIdx.x * blockDim.x + threadIdx.x;
    if (i < n4) dst[i] = make_float4(0.f, 0.f, 0.f, 0.f);
}

// ---------------------------------------------------------------------------
// KG aggregation: one wave32 per edge, lane = float2 channel pair.
// agg[head] += cur[tail] * mask * weight[type-1]
// ---------------------------------------------------------------------------
__global__ void __launch_bounds__(256)
kg_scatter(const float* __restrict__ cur,
           const float* __restrict__ weight,
           const float* __restrict__ mask,
           const int* __restrict__ head,
           const int* __restrict__ tail,
           const int* __restrict__ etype,
           float* __restrict__ agg, int nEdges) {
    int gid  = blockIdx.x * blockDim.x + threadIdx.x;
    int e    = gid >> 5;                 // wave-uniform edge id
    int lane = threadIdx.x & 31;
    if (e >= nEdges) return;             // whole-wave uniform exit

    // keep the index/mask streams hot (global_prefetch_b8)
    if (lane == 0 && e + 4096 < nEdges) {
        __builtin_prefetch(&tail[e + 4096], 0, 1);
        __builtin_prefetch(&head[e + 4096], 0, 1);
        __builtin_prefetch(&mask[e + 4096], 0, 1);
    }

    int   h  = head[e];
    int   t  = tail[e];
    int   ty = etype[e] - 1;
    float m  = mask[e];

    const float2* srow = (const float2*)(cur    + (size_t)t  * CH);
    const float2* wrow = (const float2*)(weight + (size_t)ty * CH);
    float2 v = srow[lane];               // 256B coalesced per wave
    float2 w = wrow[lane];               // L2-resident (4KB table)
    float ox = v.x * m * w.x;
    float oy = v.y * m * w.y;

    float* dst = agg + (size_t)h * CH + lane * 2;
    __hip_atomic_fetch_add(dst + 0, ox, __ATOMIC_RELAXED, __HIP_MEMORY_SCOPE_AGENT);
    __hip_atomic_fetch_add(dst + 1, oy, __ATOMIC_RELAXED, __HIP_MEMORY_SCOPE_AGENT);
}

// ---------------------------------------------------------------------------
// User aggregation (COO SpMM): one wave32 per nnz.
// agg[row] += vals * cur[col]
// ---------------------------------------------------------------------------
__global__ void __launch_bounds__(256)
user_scatter(const float* __restrict__ cur,
             const float* __restrict__ vals,
             const int* __restrict__ rows,
             const int* __restrict__ cols,
             float* __restrict__ agg, int nnz) {
    int gid  = blockIdx.x * blockDim.x + threadIdx.x;
    int i    = gid >> 5;
    int lane = threadIdx.x & 31;
    if (i >= nnz) return;

    if (lane == 0 && i + 4096 < nnz) {
        __builtin_prefetch(&cols[i + 4096], 0, 1);
        __builtin_prefetch(&rows[i + 4096], 0, 1);
    }

    int   r = rows[i];
    int   c = cols[i];
    float s = vals[i];

    const float2* srow = (const float2*)(cur + (size_t)c * CH);
    float2 v = srow[lane];
    float ox = v.x * s;
    float oy = v.y * s;

    float* dst = agg + (size_t)r * CH + lane * 2;
    __hip_atomic_fetch_add(dst + 0, ox, __ATOMIC_RELAXED, __HIP_MEMORY_SCOPE_AGENT);
    __hip_atomic_fetch_add(dst + 1, oy, __ATOMIC_RELAXED, __HIP_MEMORY_SCOPE_AGENT);
}

// ---------------------------------------------------------------------------
// L2-normalize 16-row tiles + residual accumulate, squared norms via WMMA.
//
// For a 16x64 tile X, the Gram matrix X*X^T is accumulated with 16x
// V_WMMA_F32_16X16X4_F32 (K=64 in K=4 steps). The f32 A(16x4) and B(4x16)
// VGPR layouts are lane-identical for this symmetric product, so B == A
// (same register pair). Diagonal of the result = squared row norms:
//   row m<8  -> lane m,    acc[m]
//   row m>=8 -> lane m+16, acc[m-8]
// broadcast with __shfl; no LDS, no barrier. One wave per tile, 8 tiles/block.
// ---------------------------------------------------------------------------
__global__ void __launch_bounds__(256)
normalize_accum(const float* __restrict__ in,      // [nTiles*16, 64] aggregate
                float* __restrict__ outNorm,       // normalized state (may be null)
                float* __restrict__ res,           // residual accumulator
                int nTiles) {
    int lane = threadIdx.x & 31;
    int wv   = threadIdx.x >> 5;
    int tile = blockIdx.x * 8 + wv;
    if (tile >= nTiles) return;          // wave-uniform: EXEC stays all-ones

    int r    = lane & 15;                // row within tile
    int koff = (lane >> 4) * 2;          // lanes 16..31 carry K=2,3 of each chunk
    size_t rowBase = ((size_t)tile * 16 + r) * CH;

    float2 xs[16];
    v8f acc = {};
#pragma unroll
    for (int i = 0; i < 16; ++i) {
        xs[i] = *(const float2*)(in + rowBase + 4 * i + koff);
        v2f a; a.x = xs[i].x; a.y = xs[i].y;
        // D = A * A^T + D  (Gram accumulation, fp32, RNE)
        acc = __builtin_amdgcn_wmma_f32_16x16x4_f32(
            /*neg_a=*/false, a, /*neg_b=*/false, a,
            /*c_mod=*/(short)0, acc, /*reuse_a=*/false, /*reuse_b=*/false);
    }

    // pick this lane's potential diagonal element (elem index = lane&7)
    float d;
    switch (lane & 7) {
        case 0: d = acc[0]; break;
        case 1: d = acc[1]; break;
        case 2: d = acc[2]; break;
        case 3: d = acc[3]; break;
        case 4: d = acc[4]; break;
        case 5: d = acc[5]; break;
        case 6: d = acc[6]; break;
        default: d = acc[7]; break;
    }
    // squared norm of row r lives on lane r (r<8) or lane r+16 (r>=8)
    int srcLane = r + ((r >= 8) ? 16 : 0);
    float nsq = __shfl(d, srcLane, 32);

    float nrm = sqrtf(nsq);
    float inv = 1.0f / fmaxf(nrm, 1e-12f);   // matches x / max(||x||, eps)

#pragma unroll
    for (int i = 0; i < 16; ++i) {
        size_t off = rowBase + 4 * i + koff;
        float2 y;
        y.x = xs[i].x * inv;
        y.y = xs[i].y * inv;
        if (outNorm) *(float2*)(outNorm + off) = y;    // next-hop state
        float2 rv = *(const float2*)(res + off);       // residual += y
        rv.x += y.x;
        rv.y += y.y;
        *(float2*)(res + off) = rv;
    }
}

// ---------------------------------------------------------------------------
// driver
// ---------------------------------------------------------------------------
extern "C" void kernel_launch(void* const* d_in, const int* in_sizes, int n_in,
                              void* d_out, int out_size, void* d_ws, size_t ws_size,
                              hipStream_t stream) {
    const float* user_emb   = (const float*)d_in[0];
    const float* entity_emb = (const float*)d_in[1];
    const float* weight     = (const float*)d_in[2];
    const float* tmask      = (const float*)d_in[3];
    const float* ivals      = (const float*)d_in[4];
    const int*   ehead      = (const int*)d_in[5];
    const int*   etail      = (const int*)d_in[6];
    const int*   etype      = (const int*)d_in[7];
    const int*   irows      = (const int*)d_in[8];
    const int*   icols      = (const int*)d_in[9];

    float* out     = (float*)d_out;
    float* ent_res = out;                                   // [N_ENTITIES, CH]
    float* usr_res = out + (size_t)N_ENTITIES * CH;         // [N_USERS, CH]

    float* ws      = (float*)d_ws;
    float* ent_cur = ws;                                    // 25.6 MB
    float* ent_agg = ent_cur + (size_t)N_ENTITIES * CH;     // 25.6 MB
    float* usr_agg = ent_agg + (size_t)N_ENTITIES * CH;     // 12.8 MB
    // ent_agg and usr_agg are contiguous -> single zero pass

    const int entN   = N_ENTITIES * CH;       // 6,400,000
    const int usrN   = N_USERS * CH;          // 3,200,000
    const int entN4  = entN / 4;
    const int usrN4  = usrN / 4;
    const int aggN4  = (entN + usrN) / 4;     // 2,400,000

    // init: residuals = input embeddings; ent_cur = entity_emb
    copy2_f4<<<(entN4 + 255) / 256, 256, 0, stream>>>(
        (const float4*)entity_emb, (float4*)ent_res, (float4*)ent_cur, entN4);
    copy_f4<<<(usrN4 + 255) / 256, 256, 0, stream>>>(
        (const float4*)user_emb, (float4*)usr_res, usrN4);

    const int entTiles = N_ENTITIES / 16;     // 6250
    const int usrTiles = N_USERS / 16;        // 3125

    for (int hop = 0; hop < N_HOPS; ++hop) {
        zero_f4<<<(aggN4 + 255) / 256, 256, 0, stream>>>((float4*)ent_agg, aggN4);

        kg_scatter<<<(int)(((long long)N_EDGES * 32 + 255) / 256), 256, 0, stream>>>(
            ent_cur, weight, tmask, ehead, etail, etype, ent_agg, N_EDGES);

        user_scatter<<<(int)(((long long)NNZ * 32 + 255) / 256), 256, 0, stream>>>(
            ent_cur, ivals, irows, icols, usr_agg, NNZ);

        // normalize + residual; entity writes its next-hop state in place
        normalize_accum<<<(entTiles + 7) / 8, 256, 0, stream>>>(
            ent_agg, ent_cur, ent_res, entTiles);
        normalize_accum<<<(usrTiles + 7) / 8, 256, 0, stream>>>(
            usr_agg, (float*)nullptr, usr_res, usrTiles);
    }
}